// DocREModel_12240656794093
// MI455X (gfx1250) — compile-verified
//
#include <hip/hip_runtime.h>
#include <hip/hip_bf16.h>

typedef __attribute__((ext_vector_type(2))) float v2f;
typedef __attribute__((ext_vector_type(8))) float v8f;

#define B_   2
#define L_   1024
#define H_   768
#define NH_  12
#define NE_  24
#define M_   3
#define NC_  2
#define CW_  8
#define BLK_ 64
#define NCLS_ 97
#define KB_  12          // H/BLOCK
#define HB_  49152       // H*BLOCK
#define EF_  576         // NE*NE
#define OFF_ 1

// ---------------------------------------------------------------------------
// Wave-level fp32 WMMA 16x16 tile GEMM over K (step 4), D = A(16xK) @ B(Kx16).
// A fragment (ISA 7.12.2, 32-bit A 16x4): lane holds row (lane&15),
// k-offset (lane>>4)*2 + {0,1}. B mirrors by column. C/D: 8 VGPRs.
// ---------------------------------------------------------------------------
template <bool BT>
__device__ inline v8f gemm_tile_f32(const float* __restrict__ A, int lda,
                                    const float* __restrict__ Bm, int ldb,
                                    int K, int row, int col, int koff) {
    v8f c = {};
    for (int k = 0; k < K; k += 4) {
        v2f a, b;
        a.x = A[(size_t)row * lda + k + koff];
        a.y = A[(size_t)row * lda + k + koff + 1];
        if (BT) {   // B stored as (N x K) row-major -> contiguous per-lane pair
            b.x = Bm[(size_t)col * ldb + k + koff];
            b.y = Bm[(size_t)col * ldb + k + koff + 1];
        } else {    // B stored as (K x N) row-major
            b.x = Bm[(size_t)(k + koff) * ldb + col];
            b.y = Bm[(size_t)(k + koff + 1) * ldb + col];
        }
        c = __builtin_amdgcn_wmma_f32_16x16x4_f32(false, a, false, b,
                                                  (short)0, c, false, false);
    }
    return c;
}

// ---------------------------------------------------------------------------
// K1: e_att[b,e,h,l] = mean_m attention[b,h,p_m,l];  gate[b,e,l] normalized
// ---------------------------------------------------------------------------
__global__ void k_eatt_gate(const float* __restrict__ att,
                            const int* __restrict__ mst,
                            float* __restrict__ e_att,
                            float* __restrict__ gate) {
    int be = blockIdx.x;               // b*NE + e
    int b = be / NE_;
    int p[M_];
    for (int m = 0; m < M_; ++m) p[m] = mst[be * M_ + m] + OFF_;
    float lsum = 0.f;
    for (int l = threadIdx.x; l < L_; l += blockDim.x) {
        float rowsum = 0.f;
        for (int h = 0; h < NH_; ++h) {
            float s = 0.f;
            for (int m = 0; m < M_; ++m)
                s += att[(((size_t)b * NH_ + h) * L_ + p[m]) * L_ + l];
            s *= (1.0f / M_);
            e_att[((size_t)be * NH_ + h) * L_ + l] = s;
            rowsum += s;
        }
        gate[(size_t)be * L_ + l] = rowsum;     // unnormalized for now
        lsum += rowsum;
    }
    __shared__ float sb[256];
    sb[threadIdx.x] = lsum;
    __syncthreads();
    for (int s = blockDim.x / 2; s > 0; s >>= 1) {
        if (threadIdx.x < s) sb[threadIdx.x] += sb[threadIdx.x + s];
        __syncthreads();
    }
    float inv = 1.0f / sb[0];
    for (int l = threadIdx.x; l < L_; l += blockDim.x)
        gate[(size_t)be * L_ + l] *= inv;
}

// ---------------------------------------------------------------------------
// K2: e_emb[b,e,h] = logsumexp over {3 mention rows, 2 gate-weighted windows}
// ---------------------------------------------------------------------------
__global__ void k_eemb(const float* __restrict__ seq,
                       const int* __restrict__ mst,
                       const int* __restrict__ cst,
                       const float* __restrict__ gate,
                       float* __restrict__ eemb) {
    int be = blockIdx.x;
    int b = be / NE_;
    int p[M_], cs[NC_];
    for (int m = 0; m < M_; ++m)  p[m]  = mst[be * M_ + m] + OFF_;
    for (int c = 0; c < NC_; ++c) cs[c] = cst[be * NC_ + c];
    __shared__ float g[NC_][CW_];
    if (threadIdx.x < NC_ * CW_) {
        int c = threadIdx.x / CW_, w = threadIdx.x % CW_;
        g[c][w] = gate[(size_t)be * L_ + cs[c] + w];
    }
    __syncthreads();
    for (int h = threadIdx.x; h < H_; h += blockDim.x) {
        float v[M_ + NC_];
        for (int m = 0; m < M_; ++m)
            v[m] = seq[((size_t)b * L_ + p[m]) * H_ + h];
        for (int c = 0; c < NC_; ++c) {
            float s = 0.f;
            for (int w = 0; w < CW_; ++w)
                s += g[c][w] * seq[((size_t)b * L_ + cs[c] + w) * H_ + h];
            v[M_ + c] = s;
        }
        float mx = v[0];
        for (int i = 1; i < M_ + NC_; ++i) mx = fmaxf(mx, v[i]);
        float s = 0.f;
        for (int i = 0; i < M_ + NC_; ++i) s += expf(v[i] - mx);
        eemb[(size_t)be * H_ + h] = mx + logf(s);
    }
}

// ---------------------------------------------------------------------------
// K3: ht[b,e,f,l] = relu(sum_h e_att[e,h,l]*e_att[f,h,l]) normalized over l
// ---------------------------------------------------------------------------
__global__ void k_ht(const float* __restrict__ e_att, float* __restrict__ ht) {
    int idx = blockIdx.x;              // b*EF + e*NE + f
    int b = idx / EF_, ef = idx % EF_;
    int e = ef / NE_, f = ef % NE_;
    const float* ea = e_att + ((size_t)(b * NE_ + e) * NH_) * L_;
    const float* fa = e_att + ((size_t)(b * NE_ + f) * NH_) * L_;
    float lsum = 0.f;
    for (int l = threadIdx.x; l < L_; l += blockDim.x) {
        float s = 0.f;
        for (int h = 0; h < NH_; ++h) s += ea[h * L_ + l] * fa[h * L_ + l];
        s = fmaxf(s, 0.f);
        ht[(size_t)idx * L_ + l] = s;
        lsum += s;
    }
    __shared__ float sb[256];
    sb[threadIdx.x] = lsum;
    __syncthreads();
    for (int s = blockDim.x / 2; s > 0; s >>= 1) {
        if (threadIdx.x < s) sb[threadIdx.x] += sb[threadIdx.x + s];
        __syncthreads();
    }
    float inv = 1.0f / (sb[0] + 1e-10f);
    for (int l = threadIdx.x; l < L_; l += blockDim.x)
        ht[(size_t)idx * L_ + l] *= inv;
}

// ---------------------------------------------------------------------------
// K4: rs[b] (576x768) = ht[b] (576x1024) @ seq[b] (1024x768).  1 wave / tile.
// ---------------------------------------------------------------------------
__global__ void __launch_bounds__(32) k_rs(const float* __restrict__ ht,
                                           const float* __restrict__ seq,
                                           float* __restrict__ rs) {
    int b = blockIdx.z;
    int lane = threadIdx.x & 31;
    int row = blockIdx.x * 16 + (lane & 15);
    int col = blockIdx.y * 16 + (lane & 15);
    int koff = (lane >> 4) * 2;
    const float* A  = ht  + (size_t)b * EF_ * L_;
    const float* Bm = seq + (size_t)b * L_ * H_;
    v8f c = gemm_tile_f32<false>(A, L_, Bm, H_, L_, row, col, koff);
    float* C = rs + (size_t)b * EF_ * H_;
    int rowb = blockIdx.x * 16 + (lane >> 4) * 8;
    int colc = blockIdx.y * 16 + (lane & 15);
    for (int r = 0; r < 8; ++r)
        C[(size_t)(rowb + r) * H_ + colc] = c[r];
}

// ---------------------------------------------------------------------------
// K5: he[be,d] = b_head[d] + sum_j eemb[be,j]*W_head[d,j]; same for te/W_tail
// ---------------------------------------------------------------------------
__global__ void k_hete(const float* __restrict__ eemb,
                       const float* __restrict__ Wh, const float* __restrict__ bh,
                       const float* __restrict__ Wt, const float* __restrict__ bt,
                       float* __restrict__ he, float* __restrict__ te) {
    int be = blockIdx.x;
    __shared__ float es[H_];
    for (int j = threadIdx.x; j < H_; j += blockDim.x)
        es[j] = eemb[(size_t)be * H_ + j];
    __syncthreads();
    for (int d = threadIdx.x; d < H_; d += blockDim.x) {
        float sh = bh[d], st = bt[d];
        const float* wh = Wh + (size_t)d * 2 * H_;
        const float* wt = Wt + (size_t)d * 2 * H_;
        for (int j = 0; j < H_; ++j) {
            sh += es[j] * wh[j];
            st += es[j] * wt[j];
        }
        he[(size_t)be * H_ + d] = sh;
        te[(size_t)be * H_ + d] = st;
    }
}

// ---------------------------------------------------------------------------
// K6: z[b,ef,d] = tanh(ebias[b, e|f, d] + sum_j rs[b,ef,j]*W[d, H+j])
//     (transposed-B access into W's second half)
// ---------------------------------------------------------------------------
__global__ void __launch_bounds__(32) k_z(const float* __restrict__ rs,
                                          const float* __restrict__ W,
                                          const float* __restrict__ ebias,
                                          int tail, float* __restrict__ z) {
    int b = blockIdx.z;
    int lane = threadIdx.x & 31;
    int row = blockIdx.x * 16 + (lane & 15);
    int col = blockIdx.y * 16 + (lane & 15);
    int koff = (lane >> 4) * 2;
    const float* A = rs + (size_t)b * EF_ * H_;
    v8f c = gemm_tile_f32<true>(A, H_, W + H_, 2 * H_, H_, row, col, koff);
    float* C = z + (size_t)b * EF_ * H_;
    int rowb = blockIdx.x * 16 + (lane >> 4) * 8;
    int colc = blockIdx.y * 16 + (lane & 15);
    for (int r = 0; r < 8; ++r) {
        int rr = rowb + r;
        int e = rr / NE_, f = rr % NE_;
        int eb = tail ? f : e;
        float v = c[r] + ebias[(size_t)(b * NE_ + eb) * H_ + colc];
        C[(size_t)rr * H_ + colc] = tanhf(v);
    }
}

// ---------------------------------------------------------------------------
// K7: Wc (97 x 49152) = W_cls (97x768) @ W_proj (768x49152)
// ---------------------------------------------------------------------------
__global__ void __launch_bounds__(32) k_wc(const float* __restrict__ Wcls,
                                           const float* __restrict__ Wproj,
                                           float* __restrict__ Wc) {
    int lane = threadIdx.x & 31;
    int row = blockIdx.y * 16 + (lane & 15);
    if (row > NCLS_ - 1) row = NCLS_ - 1;          // clamp loads (no divergence)
    int col = blockIdx.x * 16 + (lane & 15);
    int koff = (lane >> 4) * 2;
    v8f c = gemm_tile_f32<false>(Wcls, H_, Wproj, HB_, H_, row, col, koff);
    int rowb = blockIdx.y * 16 + (lane >> 4) * 8;
    int colc = blockIdx.x * 16 + (lane & 15);
    for (int r = 0; r < 8; ++r) {
        int rr = rowb + r;
        if (rr < NCLS_) Wc[(size_t)rr * HB_ + colc] = c[r];
    }
}

// ---------------------------------------------------------------------------
// K8: logits[pair,c] = b_cls[c] + sum_{k,i,j} Wc[c,k*4096+i*64+j]
//                      * zh[pair,k*64+i] * zt[pair,k*64+j]
// A-fragment built on the fly from 8 KB LDS stage per 64-wide k-block;
// never materializes the 226 MB `bl` tensor.
// ---------------------------------------------------------------------------
__global__ void __launch_bounds__(32) k_logits(const float* __restrict__ zh,
                                               const float* __restrict__ zt,
                                               const float* __restrict__ Wc,
                                               const float* __restrict__ bcls,
                                               float* __restrict__ out) {
    int pt = blockIdx.x;               // pair tile (16 pairs)
    int ct = blockIdx.y;               // class tile
    int lane = threadIdx.x & 31;
    int mrow = lane & 15;
    int koff = (lane >> 4) * 2;
    int col = ct * 16 + (lane & 15);
    int colc = (col < NCLS_) ? col : (NCLS_ - 1);   // clamp loads

    __shared__ float zhs[16 * BLK_];
    __shared__ float zts[16 * BLK_];
    v8f c = {};

    for (int kb = 0; kb < KB_; ++kb) {
        __syncthreads();
        for (int i = threadIdx.x; i < 16 * BLK_; i += blockDim.x) {
            int r = i >> 6, d = i & 63;
            size_t g = (size_t)(pt * 16 + r) * H_ + kb * BLK_ + d;
            zhs[i] = zh[g];
            zts[i] = zt[g];
        }
        __syncthreads();
        const float* wrow = Wc + (size_t)colc * HB_ + (size_t)kb * 4096;
        for (int q = 0; q < 4096; q += 4) {
            int q0 = q + koff, q1 = q0 + 1;
            int i0 = q0 >> 6, j0 = q0 & 63;
            int i1 = q1 >> 6, j1 = q1 & 63;
            v2f a, b;
            a.x = zhs[mrow * BLK_ + i0] * zts[mrow * BLK_ + j0];
            a.y = zhs[mrow * BLK_ + i1] * zts[mrow * BLK_ + j1];
            b.x = wrow[q0];
            b.y = wrow[q1];
            c = __builtin_amdgcn_wmma_f32_16x16x4_f32(false, a, false, b,
                                                      (short)0, c, false, false);
        }
    }
    int rowb = pt * 16 + (lane >> 4) * 8;
    int cc = ct * 16 + (lane & 15);
    if (cc < NCLS_) {
        float bc = bcls[cc];
        for (int r = 0; r < 8; ++r)
            out[(size_t)(rowb + r) * NCLS_ + cc] = c[r] + bc;
    }
}

// ---------------------------------------------------------------------------
extern "C" void kernel_launch(void* const* d_in, const int* in_sizes, int n_in,
                              void* d_out, int out_size, void* d_ws, size_t ws_size,
                              hipStream_t stream) {
    (void)in_sizes; (void)n_in; (void)out_size; (void)ws_size;
    const float* seq   = (const float*)d_in[0];   // (B,L,H)
    const float* att   = (const float*)d_in[1];   // (B,NH,L,L)
    const int*   mst   = (const int*)d_in[2];     // (B,NE,M)
    const int*   cst   = (const int*)d_in[3];     // (B,NE,NC)
    const float* Wh    = (const float*)d_in[4];   // (H,2H)
    const float* bh    = (const float*)d_in[5];   // (H)
    const float* Wt    = (const float*)d_in[6];   // (H,2H)
    const float* bt    = (const float*)d_in[7];   // (H)
    const float* Wproj = (const float*)d_in[8];   // (H,H*BLOCK)
    const float* Wcls  = (const float*)d_in[9];   // (NCLS,H)
    const float* bcls  = (const float*)d_in[10];  // (NCLS)
    float* out = (float*)d_out;                   // (B,NE,NE,NCLS)

    float* ws = (float*)d_ws;
    size_t o = 0;
    float* e_att = ws + o; o += (size_t)B_ * NE_ * NH_ * L_;   // 589824
    float* gate  = ws + o; o += (size_t)B_ * NE_ * L_;          // 49152
    float* eemb  = ws + o; o += (size_t)B_ * NE_ * H_;          // 36864
    float* ht    = ws + o; o += (size_t)B_ * EF_ * L_;          // 1179648
    float* rs    = ws + o; o += (size_t)B_ * EF_ * H_;          // 884736
    float* he    = ws + o; o += (size_t)B_ * NE_ * H_;
    float* te    = ws + o; o += (size_t)B_ * NE_ * H_;
    float* zh    = ws + o; o += (size_t)B_ * EF_ * H_;
    float* zt    = ws + o; o += (size_t)B_ * EF_ * H_;
    float* Wc    = ws + o; o += (size_t)NCLS_ * HB_;            // 4767744
    // total ~9.35M floats (~37.4 MB)

    k_eatt_gate<<<B_ * NE_, 256, 0, stream>>>(att, mst, e_att, gate);
    k_eemb<<<B_ * NE_, 256, 0, stream>>>(seq, mst, cst, gate, eemb);
    k_ht<<<B_ * EF_, 256, 0, stream>>>(e_att, ht);
    k_rs<<<dim3(EF_ / 16, H_ / 16, B_), 32, 0, stream>>>(ht, seq, rs);
    k_hete<<<B_ * NE_, 256, 0, stream>>>(eemb, Wh, bh, Wt, bt, he, te);
    k_z<<<dim3(EF_ / 16, H_ / 16, B_), 32, 0, stream>>>(rs, Wh, he, 0, zh);
    k_z<<<dim3(EF_ / 16, H_ / 16, B_), 32, 0, stream>>>(rs, Wt, te, 1, zt);
    k_wc<<<dim3(HB_ / 16, (NCLS_ + 15) / 16), 32, 0, stream>>>(Wcls, Wproj, Wc);
    k_logits<<<dim3((B_ * EF_) / 16, (NCLS_ + 15) / 16), 32, 0, stream>>>(
        zh, zt, Wc, bcls, out);
}